// GroupBInternalPipeline_78288663872284
// MI455X (gfx1250) — compile-verified
//
#include <hip/hip_runtime.h>
#include <math.h>

#define BB   512
#define NN   128
#define DD   256
#define HH   8
#define HDIM 32
#define KSEL 4
#define BHID 64

#define INV_SCALE 0.17677669529663687f   // 1/sqrt(32)

typedef __attribute__((ext_vector_type(2))) float v2f;
typedef __attribute__((ext_vector_type(8))) float v8f;

// ---------------------------------------------------------------------------
// K1: per-batch ego token: LN1 -> Ve = ego@cWv -> ov = Ve@cWo
// (softmax over a singleton axis == 1, so out1 row == ov for every n)
// ---------------------------------------------------------------------------
__global__ __launch_bounds__(DD) void ego_kernel(
    const float* __restrict__ tokens, const int* __restrict__ ego_idx,
    const float* __restrict__ ln1_g, const float* __restrict__ ln1_b,
    const float* __restrict__ cWv, const float* __restrict__ cWo,
    float* __restrict__ ov) {
  int b = blockIdx.x, t = threadIdx.x;
  __shared__ float red[DD];
  __shared__ float sEgo[DD];
  __shared__ float sVe[DD];
  int eidx = ego_idx[b];
  float v = tokens[((long)b * NN + eidx) * DD + t];
  red[t] = v; __syncthreads();
  for (int s = DD / 2; s > 0; s >>= 1) { if (t < s) red[t] += red[t + s]; __syncthreads(); }
  float mu = red[0] * (1.0f / DD); __syncthreads();
  float dv = v - mu;
  red[t] = dv * dv; __syncthreads();
  for (int s = DD / 2; s > 0; s >>= 1) { if (t < s) red[t] += red[t + s]; __syncthreads(); }
  float var = red[0] * (1.0f / DD);
  sEgo[t] = dv * rsqrtf(var + 1e-5f) * ln1_g[t] + ln1_b[t];
  __syncthreads();
  float acc = 0.f;
  for (int i = 0; i < DD; ++i) acc += sEgo[i] * cWv[i * DD + t];
  sVe[t] = acc; __syncthreads();
  float acc2 = 0.f;
  for (int i = 0; i < DD; ++i) acc2 += sVe[i] * cWo[i * DD + t];
  ov[b * DD + t] = acc2;
}

// ---------------------------------------------------------------------------
// K2: h1 = tokens + ov (broadcast over n), write h1 -> d_out; x2 = LN2(h1) -> ws
// ---------------------------------------------------------------------------
__global__ __launch_bounds__(DD) void h1_ln2_kernel(
    const float* __restrict__ tokens, const float* __restrict__ ov,
    const float* __restrict__ ln2_g, const float* __restrict__ ln2_b,
    float* __restrict__ h1_out, float* __restrict__ x2) {
  long row = blockIdx.x;
  int t = threadIdx.x;
  int b = (int)(row / NN);
  float h = tokens[row * DD + t] + ov[b * DD + t];
  h1_out[row * DD + t] = h;
  __shared__ float red[DD];
  red[t] = h; __syncthreads();
  for (int s = DD / 2; s > 0; s >>= 1) { if (t < s) red[t] += red[t + s]; __syncthreads(); }
  float mu = red[0] * (1.0f / DD); __syncthreads();
  float dv = h - mu;
  red[t] = dv * dv; __syncthreads();
  for (int s = DD / 2; s > 0; s >>= 1) { if (t < s) red[t] += red[t + s]; __syncthreads(); }
  float var = red[0] * (1.0f / DD);
  x2[row * DD + t] = dv * rsqrtf(var + 1e-5f) * ln2_g[t] + ln2_b[t];
}

// ---------------------------------------------------------------------------
// K3: per-batch 5 smallest distances (stable, matches top_k tie-break) and
//     Kc/Vc = x2[b, top5] @ pWk / pWv  (only 5 distinct neighbor rows/batch)
// ---------------------------------------------------------------------------
__global__ __launch_bounds__(DD) void top5_kv_kernel(
    const float* __restrict__ ego_distances, const float* __restrict__ x2,
    const float* __restrict__ pWk, const float* __restrict__ pWv,
    int* __restrict__ top5, float* __restrict__ Kc, float* __restrict__ Vc) {
  int b = blockIdx.x, t = threadIdx.x;
  __shared__ int sIdx[5];
  __shared__ float sX[5 * DD];
  if (t == 0) {
    for (int r = 0; r < 5; ++r) {
      float best = INFINITY; int bi = 0;
      for (int j = 0; j < NN; ++j) {
        bool taken = false;
        for (int p = 0; p < r; ++p) if (sIdx[p] == j) taken = true;
        float d = ego_distances[b * NN + j];
        if (!taken && d < best) { best = d; bi = j; }
      }
      sIdx[r] = bi;
      top5[b * 5 + r] = bi;
    }
  }
  __syncthreads();
  for (int r = 0; r < 5; ++r)
    sX[r * DD + t] = x2[((long)b * NN + sIdx[r]) * DD + t];
  __syncthreads();
  for (int r = 0; r < 5; ++r) {
    float ak = 0.f, av = 0.f;
    for (int i = 0; i < DD; ++i) {
      float xv = sX[r * DD + i];
      ak += xv * pWk[i * DD + t];
      av += xv * pWv[i * DD + t];
    }
    Kc[(b * 5 + r) * DD + t] = ak;
    Vc[(b * 5 + r) * DD + t] = av;
  }
}

// ---------------------------------------------------------------------------
// WMMA fp32 GEMM: out[M,256] = A[M,256] @ W[256,256] (+ addend), via
// V_WMMA_F32_16X16X4_F32. Block tile 128x64, 8 waves, each wave 16x64.
// Fragment lane mapping per CDNA5 ISA 7.12.2:
//   A (16x4):  lane l -> M=l%16, K = vgpr + 2*(l/16)
//   B (4x16):  lane l -> N=l%16, K = vgpr + 2*(l/16)
//   C (16x16): lane l, vgpr v -> N=l%16, M = v + 8*(l/16)
// ---------------------------------------------------------------------------
#define BM 128
#define BN 64
#define BK 16

__global__ __launch_bounds__(256) void gemm_wmma_f32(
    const float* __restrict__ A, const float* __restrict__ W,
    const float* __restrict__ addend, float* __restrict__ out) {
  __shared__ float As[BM][BK + 1];   // +1 pad: conflict-free A-frag reads
  __shared__ float Bs[BK][BN + 16];  // +16 pad: keeps the two half-wave groups on disjoint banks
  int tid = threadIdx.x;
  int wave = tid >> 5;
  int lane = tid & 31;
  int laneM = lane & 15;
  int kh = lane >> 4;          // which K-pair this half-wave holds
  int blockM = blockIdx.x * BM;
  int blockN = blockIdx.y * BN;

  v8f zero = {0.f, 0.f, 0.f, 0.f, 0.f, 0.f, 0.f, 0.f};
  v8f c[4];
#pragma unroll
  for (int i = 0; i < 4; ++i) c[i] = zero;

  int acol = tid & 15, arow0 = tid >> 4;   // A staging: 16 rows / pass
  int bcol = tid & 63, brow0 = tid >> 6;   // B staging: 4 rows / pass

  for (int kc = 0; kc < DD; kc += BK) {
#pragma unroll
    for (int p = 0; p < 8; ++p) {
      int ar = arow0 + p * 16;
      As[ar][acol] = A[(long)(blockM + ar) * DD + kc + acol];
    }
#pragma unroll
    for (int p = 0; p < 4; ++p) {
      int br = brow0 + p * 4;
      Bs[br][bcol] = W[(long)(kc + br) * DD + blockN + bcol];
    }
    if (kc + BK < DD) {  // prefetch next A panel -> global_prefetch_b8
      __builtin_prefetch(&A[(long)(blockM + arow0) * DD + kc + BK + acol], 0, 1);
    }
    __syncthreads();
#pragma unroll
    for (int ks = 0; ks < 4; ++ks) {
      int k0 = ks * 4 + kh * 2;
      v2f a;
      a.x = As[wave * 16 + laneM][k0];
      a.y = As[wave * 16 + laneM][k0 + 1];
#pragma unroll
      for (int nt = 0; nt < 4; ++nt) {
        v2f bf;
        bf.x = Bs[k0][nt * 16 + laneM];
        bf.y = Bs[k0 + 1][nt * 16 + laneM];
        c[nt] = __builtin_amdgcn_wmma_f32_16x16x4_f32(
            false, a, false, bf, (short)0, c[nt], false, false);
      }
    }
    __syncthreads();
  }

#pragma unroll
  for (int nt = 0; nt < 4; ++nt) {
#pragma unroll
    for (int v = 0; v < 8; ++v) {
      int row = blockM + wave * 16 + v + kh * 8;
      int col = blockN + nt * 16 + laneM;
      long off = (long)row * DD + col;
      float val = c[nt][v];
      if (addend) val += addend[off];
      out[off] = val;
    }
  }
}

// ---------------------------------------------------------------------------
// K4b: overwrite the 512 ego rows with x2 @ eWq
// ---------------------------------------------------------------------------
__global__ __launch_bounds__(DD) void ego_q_fixup(
    const float* __restrict__ x2, const int* __restrict__ ego_idx,
    const float* __restrict__ eWq, float* __restrict__ Qp) {
  int b = blockIdx.x, t = threadIdx.x;
  __shared__ float sX[DD];
  long row = (long)b * NN + ego_idx[b];
  sX[t] = x2[row * DD + t];
  __syncthreads();
  float acc = 0.f;
  for (int i = 0; i < DD; ++i) acc += sX[i] * eWq[i * DD + t];
  Qp[row * DD + t] = acc;
}

// ---------------------------------------------------------------------------
// K5: fused attention: scores (8 heads x 4 neighbors) + distance-bias MLP +
//     softmax over K + weighted V sum. One wave per token row; lane = (h,k).
//     Reads Qp rows, overwrites them in place with attn.
// ---------------------------------------------------------------------------
__global__ __launch_bounds__(256) void attn_kernel(
    const float* __restrict__ ego_distances, const int* __restrict__ top5,
    const float* __restrict__ Kc, const float* __restrict__ Vc,
    const float* __restrict__ bW1, const float* __restrict__ bb1,
    const float* __restrict__ bW2, const float* __restrict__ bb2,
    float* __restrict__ QpAttn) {
  int tid = threadIdx.x, wave = tid >> 5, lane = tid & 31;
  long rowBase = (long)blockIdx.x * 8;
  int b = (int)(rowBase / NN);
  __shared__ float sQ[8 * DD];
  __shared__ float sKc[5 * DD];
  __shared__ float sVc[5 * DD];
  __shared__ float sW[8][HH][KSEL];
  __shared__ int sSel[8][KSEL];
  __shared__ int sT5[5];

  for (int i = tid; i < 8 * DD; i += 256) sQ[i] = QpAttn[rowBase * DD + i];
  for (int i = tid; i < 5 * DD; i += 256) {
    sKc[i] = Kc[(long)b * 5 * DD + i];
    sVc[i] = Vc[(long)b * 5 * DD + i];
  }
  if (tid < 5) sT5[tid] = top5[b * 5 + tid];
  __syncthreads();

  int n = (int)(rowBase % NN) + wave;
  int h = lane >> 2, k = lane & 3;

  // k-th top-5 candidate excluding self (matches top_k with self masked to inf)
  int sel = 0, cnt = 0;
  for (int r = 0; r < 5; ++r) {
    int idx = sT5[r];
    if (idx != n) { if (cnt == k) sel = r; ++cnt; }
  }

  float s = 0.f;
  for (int d = 0; d < HDIM; ++d)
    s += sQ[wave * DD + h * HDIM + d] * sKc[sel * DD + h * HDIM + d];
  s *= INV_SCALE;

  float d_i = ego_distances[b * NN + n];
  float d_j = ego_distances[b * NN + sT5[sel]];
  float bias = bb2[h];
  for (int j = 0; j < BHID; ++j) {
    float hv = d_i * bW1[j] + d_j * bW1[BHID + j] + bb1[j];
    hv = fmaxf(hv, 0.f);
    bias += hv * bW2[j * HH + h];
  }
  s += bias;

  // softmax across the 4 lanes sharing h (lanes h*4 .. h*4+3)
  float m = s;
  m = fmaxf(m, __shfl_xor(m, 1, 32));
  m = fmaxf(m, __shfl_xor(m, 2, 32));
  float e = expf(s - m);
  float sum = e;
  sum += __shfl_xor(sum, 1, 32);
  sum += __shfl_xor(sum, 2, 32);
  sW[wave][h][k] = e / sum;
  if (h == 0) sSel[wave][k] = sel;
  __syncthreads();

  for (int i = tid; i < 8 * DD; i += 256) {
    int w = i / DD, e2 = i % DD, hh = e2 / HDIM;
    float acc = 0.f;
#pragma unroll
    for (int kk = 0; kk < KSEL; ++kk)
      acc += sW[w][hh][kk] * sVc[sSel[w][kk] * DD + e2];
    QpAttn[rowBase * DD + i] = acc;
  }
}

// ---------------------------------------------------------------------------
extern "C" void kernel_launch(void* const* d_in, const int* in_sizes, int n_in,
                              void* d_out, int out_size, void* d_ws, size_t ws_size,
                              hipStream_t stream) {
  (void)in_sizes; (void)n_in; (void)out_size; (void)ws_size;
  const float* tokens        = (const float*)d_in[0];
  const int*   ego_idx       = (const int*)d_in[1];
  const float* ego_distances = (const float*)d_in[2];
  // d_in[3] ego_speed: unused by reference
  const float* ln1_g = (const float*)d_in[4];
  const float* ln1_b = (const float*)d_in[5];
  const float* ln2_g = (const float*)d_in[6];
  const float* ln2_b = (const float*)d_in[7];
  // d_in[8] cWq, d_in[9] cWk: dead (softmax over singleton axis)
  const float* cWv = (const float*)d_in[10];
  const float* cWo = (const float*)d_in[11];
  const float* pWq = (const float*)d_in[12];
  const float* eWq = (const float*)d_in[13];
  const float* pWk = (const float*)d_in[14];
  const float* pWv = (const float*)d_in[15];
  const float* pWo = (const float*)d_in[16];
  const float* bW1 = (const float*)d_in[17];
  const float* bb1 = (const float*)d_in[18];
  const float* bW2 = (const float*)d_in[19];
  const float* bb2 = (const float*)d_in[20];
  float* out = (float*)d_out;

  float* ws = (float*)d_ws;
  float* ov = ws;                            // 512*256
  float* x2 = ov + BB * DD;                  // 65536*256
  float* qp = x2 + (long)BB * NN * DD;       // 65536*256 (Qp, then attn in place)
  float* kc = qp + (long)BB * NN * DD;       // 512*5*256
  float* vc = kc + (long)BB * 5 * DD;        // 512*5*256
  int*  top5 = (int*)(vc + (long)BB * 5 * DD);

  ego_kernel<<<BB, DD, 0, stream>>>(tokens, ego_idx, ln1_g, ln1_b, cWv, cWo, ov);
  h1_ln2_kernel<<<BB * NN, DD, 0, stream>>>(tokens, ov, ln2_g, ln2_b, out, x2);
  top5_kv_kernel<<<BB, DD, 0, stream>>>(ego_distances, x2, pWk, pWv, top5, kc, vc);

  dim3 ggrid(BB * NN / BM, DD / BN);
  gemm_wmma_f32<<<ggrid, 256, 0, stream>>>(x2, pWq, nullptr, qp);
  ego_q_fixup<<<BB, DD, 0, stream>>>(x2, ego_idx, eWq, qp);
  attn_kernel<<<BB * NN / 8, 256, 0, stream>>>(ego_distances, top5, kc, vc,
                                               bW1, bb1, bW2, bb2, qp);
  gemm_wmma_f32<<<ggrid, 256, 0, stream>>>(qp, pWo, out, out);
}